// SpatialRationalResampler_79070347919857
// MI455X (gfx1250) — compile-verified
//
#include <hip/hip_runtime.h>

// ---------------------------------------------------------------------------
// Fused: 3x3 conv (C=128 -> 9*128) + pixel-shuffle x3 + depthwise 5x5 blur
// stride 2, for MI455X (gfx1250, wave32, WMMA f32 16x16x4).
// 27 waves: wave = (tap 0..8, third 0..2); each wave owns 7 N-tiles.
// Software-pipelined implicit GEMM: A double-buffered + global_prefetch_b8
// ahead, B-fragment LDS loads ~1.5 steps ahead, im2col offsets via LDS table.
//
// Shapes: x (2,128,8,64,64), w (1152,128,3,3), b (1152), out (2,128,8,96,96)
// ---------------------------------------------------------------------------

typedef float v2f __attribute__((ext_vector_type(2)));
typedef float v8f __attribute__((ext_vector_type(8)));
typedef int   v2i __attribute__((ext_vector_type(2)));

#define C_CH   128
#define NUMSQ  9
#define KDIM   (C_CH * NUMSQ)   // 1152
#define CR     18               // conv-pixel region per tile (16 core + 1 halo each side)
#define XR     20               // staged input region (CR + conv halo)
#define NPOS   (CR * CR)        // 324
#define CS_STRIDE 328           // padded row stride for conv-result LDS
#define THREADS 864             // 27 waves of 32
#define NTPW   7                // N-tiles per wave (3 * 7 = 21)
#define XS_FLOATS (C_CH * XR * XR)   // 51200

__device__ __forceinline__ int im2col_off_bytes(int kq) {
    int ic = kq / 9, kt = kq - ic * 9, kh = kt / 3;
    return (ic * (XR * XR) + kh * XR + (kt - kh * 3)) << 2;
}

__global__ __launch_bounds__(THREADS)
void SpatialRationalResampler_fused(const float* __restrict__ x,
                                    const float* __restrict__ w,
                                    const float* __restrict__ bias,
                                    float* __restrict__ out)
{
    extern __shared__ float smem_raw[];
    float* smem    = smem_raw;                       // input tile, later conv result
    float* bias_s  = smem_raw + XS_FLOATS;           // [144]
    v2i*   pairtab = (v2i*)(smem_raw + XS_FLOATS + 144);  // [1152] byte-offset pairs

    const int tid  = threadIdx.x;
    const int lane = tid & 31;
    const int wave = tid >> 5;       // 0..26
    const int tap  = wave / 3;       // 0..8
    const int third = wave - tap * 3;

    const int stile = blockIdx.x;    // 0..15 : 4x4 spatial tiles
    const int cg    = blockIdx.y;    // 0..7  : base-channel group of 16
    const int bf    = blockIdx.z;    // 0..15 : image index (b*8 + f)
    const int ts_h  = stile >> 2, ts_w = stile & 3;
    const int h0    = ts_h * 16,  w0   = ts_w * 16;
    const int b     = bf >> 3,    fp   = bf & 7;

    // -------- stage input tile (zero-padded) into LDS, 2 floats at a time ----
    for (int q = tid; q < XS_FLOATS / 2; q += THREADS) {
        int ic = q / 200;                  // 200 float2 per channel (400/2)
        int r2 = q - ic * 200;
        int rr = r2 / 10;                  // 10 float2 per row
        int c2 = (r2 - rr * 10) << 1;      // starting col in XR row: 0,2,..,18
        int hg  = h0 - 2 + rr;
        int wgb = w0 - 2 + c2;             // always even
        v2f v; v.x = 0.f; v.y = 0.f;
        if (hg >= 0 && hg < 64) {
            const float* src = x + (((size_t)(b * C_CH + ic) * 8 + fp) * 64 + hg) * 64 + wgb;
            if (wgb >= 0 && wgb + 1 < 64) {
                v = *(const v2f*)src;      // 8-byte aligned fast path
            } else {
                if (wgb >= 0 && wgb < 64)         v.x = src[0];
                if (wgb + 1 >= 0 && wgb + 1 < 64) v.y = src[1];
            }
        }
        *(v2f*)(smem + (q << 1)) = v;
    }
    for (int i = tid; i < 144; i += THREADS) {
        int t = i >> 4, m = i & 15;
        bias_s[i] = bias[(cg * 16 + m) * NUMSQ + t];
    }
    // im2col byte-offset pair table: pairtab[kq] = { off(kq), off(kq+1) }
    // (B 4x16 layout: vgpr0 = row k+2*khi, vgpr1 = row k+2*khi+1)
    for (int kq = tid; kq < KDIM; kq += THREADS) {
        int kq1 = (kq + 1 < KDIM) ? kq + 1 : KDIM - 1;
        v2i p; p.x = im2col_off_bytes(kq); p.y = im2col_off_bytes(kq1);
        pairtab[kq] = p;
    }
    __syncthreads();

    // -------- per-lane constants for WMMA fragments --------
    const int n16 = lane & 15;
    const int khi = lane >> 4;
    const int k2h = khi << 1;        // 0 or 2 : K-row base within a step

    int baseoff[NTPW];               // byte offsets (r*20+c)*4 for this wave's tiles
#pragma unroll
    for (int j = 0; j < NTPW; ++j) {
        int nt  = third * NTPW + j;  // thirds cover [0..6], [7..13], [14..20]
        int pos = nt * 16 + n16;
        int p   = (pos < NPOS) ? pos : 0;
        int r   = p / CR;
        int c   = p - r * CR;
        baseoff[j] = (r * XR + c) << 2;
    }

    const int ocm = (cg * 16 + n16) * NUMSQ + tap;
    const float* wp = w + (size_t)ocm * KDIM + k2h;   // A rows k+2khi, k+2khi+1
    const char*  sb = (const char*)smem;

    v8f acc[NTPW];
    const v8f vzero = {0.f, 0.f, 0.f, 0.f, 0.f, 0.f, 0.f, 0.f};
#pragma unroll
    for (int j = 0; j < NTPW; ++j) acc[j] = vzero;

    // -------- pipelined K loop (2 steps / iter, A double-buffered + prefetch)
    v2f aC0 = *(const v2f*)(wp + 0);
    v2f aC1 = *(const v2f*)(wp + 4);
    v2i op0  = pairtab[0  + k2h];
    v2i op1  = pairtab[4  + k2h];
    v2i opH0 = pairtab[8  + k2h];
    v2i opH1 = pairtab[12 + k2h];

    v2f bfA[NTPW], bfB[NTPW];
#pragma unroll
    for (int j = 0; j < NTPW; ++j) {
        bfA[j].x = *(const float*)(sb + op0.x + baseoff[j]);
        bfA[j].y = *(const float*)(sb + op0.y + baseoff[j]);
    }
#pragma unroll
    for (int j = 0; j < NTPW; ++j) {
        bfB[j].x = *(const float*)(sb + op1.x + baseoff[j]);
        bfB[j].y = *(const float*)(sb + op1.y + baseoff[j]);
    }

#pragma unroll 1
    for (int k = 0; k < KDIM; k += 8) {
        // warm L0 for the A rows ~4 iterations ahead (no counters, can't be sunk)
        __builtin_prefetch((const void*)(wp + k + 32), 0, 3);

        // prefetch A one iteration ahead, offset-pairs two iterations ahead
        int kA0 = k + 8;  if (kA0 > KDIM - 4) kA0 = KDIM - 4;
        int kA1 = k + 12; if (kA1 > KDIM - 4) kA1 = KDIM - 4;
        int kO0 = k + 16; if (kO0 > KDIM - 4) kO0 = KDIM - 4;
        int kO1 = k + 20; if (kO1 > KDIM - 4) kO1 = KDIM - 4;
        v2f aN0 = *(const v2f*)(wp + kA0);
        v2f aN1 = *(const v2f*)(wp + kA1);
        v2i opP0 = pairtab[kO0 + k2h];
        v2i opP1 = pairtab[kO1 + k2h];

        // step k : consume bfA, then refill it for step k+8
#pragma unroll
        for (int j = 0; j < NTPW; ++j)
            acc[j] = __builtin_amdgcn_wmma_f32_16x16x4_f32(
                false, aC0, false, bfA[j], (short)0, acc[j], false, false);
#pragma unroll
        for (int j = 0; j < NTPW; ++j) {
            bfA[j].x = *(const float*)(sb + opH0.x + baseoff[j]);
            bfA[j].y = *(const float*)(sb + opH0.y + baseoff[j]);
        }

        // step k+4 : consume bfB, then refill it for step k+12
#pragma unroll
        for (int j = 0; j < NTPW; ++j)
            acc[j] = __builtin_amdgcn_wmma_f32_16x16x4_f32(
                false, aC1, false, bfB[j], (short)0, acc[j], false, false);
#pragma unroll
        for (int j = 0; j < NTPW; ++j) {
            bfB[j].x = *(const float*)(sb + opH1.x + baseoff[j]);
            bfB[j].y = *(const float*)(sb + opH1.y + baseoff[j]);
        }

        aC0 = aN0; aC1 = aN1;
        opH0 = opP0; opH1 = opP1;
    }

    // -------- write conv results (+bias) into LDS (aliases input region) --------
    __syncthreads();   // everyone done reading staged input
#pragma unroll
    for (int j = 0; j < NTPW; ++j) {
        int nt  = third * NTPW + j;
        int pos = nt * 16 + n16;
        if (pos < NPOS) {
#pragma unroll
            for (int q = 0; q < 8; ++q) {
                int m = q + khi * 8;
                smem[(tap * 16 + m) * CS_STRIDE + pos] = acc[j][q] + bias_s[tap * 16 + m];
            }
        }
    }
    __syncthreads();

    // -------- pixel-shuffle + 5x5 binomial blur (stride 2, pad 2) --------
    const float k5[5] = {1.f, 4.f, 6.f, 4.f, 1.f};
    for (int i = tid; i < 16 * 24 * 24; i += THREADS) {
        int ch  = i / 576;
        int rem = i - ch * 576;
        int hl  = rem / 24;
        int wl  = rem - hl * 24;
        int hog = ts_h * 24 + hl;
        int wog = ts_w * 24 + wl;

        int  coff[5]; int tcol[5]; bool vok[5];
#pragma unroll
        for (int dx = 0; dx < 5; ++dx) {
            int v = 2 * wog + dx - 2;
            vok[dx] = (v >= 0) && (v < 192);
            int wb = v / 3;
            int p2 = v - wb * 3;
            coff[dx] = wb - (w0 - 1);
            tcol[dx] = p2;
        }

        float sum = 0.f;
#pragma unroll
        for (int dy = 0; dy < 5; ++dy) {
            int u = 2 * hog + dy - 2;
            if (u < 0 || u >= 192) continue;
            int hb = u / 3, p1 = u - hb * 3;
            int r  = hb - (h0 - 1);
            float rowsum = 0.f;
#pragma unroll
            for (int dx = 0; dx < 5; ++dx) {
                if (!vok[dx]) continue;
                int tp = p1 * 3 + tcol[dx];
                rowsum += k5[dx] * smem[(tp * 16 + ch) * CS_STRIDE + r * CR + coff[dx]];
            }
            sum += k5[dy] * rowsum;
        }
        out[(((size_t)(b * C_CH + cg * 16 + ch) * 8 + fp) * 96 + hog) * 96 + wog] =
            sum * (1.0f / 256.0f);
    }
}

extern "C" void kernel_launch(void* const* d_in, const int* in_sizes, int n_in,
                              void* d_out, int out_size, void* d_ws, size_t ws_size,
                              hipStream_t stream) {
    (void)in_sizes; (void)n_in; (void)d_ws; (void)ws_size; (void)out_size;
    const float* x    = (const float*)d_in[0];
    const float* cw   = (const float*)d_in[1];
    const float* cb   = (const float*)d_in[2];
    float*       outp = (float*)d_out;

    dim3 grid(16, 8, 16);   // spatial tiles, channel groups, images
    dim3 block(THREADS);
    // input stage (204800 B) + bias (576 B) + offset-pair table (9216 B)
    size_t shmem = (size_t)(XS_FLOATS + 144) * sizeof(float) + (size_t)KDIM * sizeof(v2i);
    hipLaunchKernelGGL(SpatialRationalResampler_fused, grid, block, shmem, stream,
                       x, cw, cb, outp);
}